// RegressionModel_65481071398593
// MI455X (gfx1250) — compile-verified
//
#include <hip/hip_runtime.h>
#include <hip/hip_bf16.h>

typedef __attribute__((ext_vector_type(2))) float v2f;
typedef __attribute__((ext_vector_type(8))) float v8f;

#define NW 12
#define QDIM 4096

// ---------------------------------------------------------------------------
// Setup: build 3x (16x16 complex) group gate matrices = Kron of 4 per-wire
// U_i = RY(phi)*RX(theta), plus folded readout vector zW[4096].
// ws layout (floats): g in 0..2: [g*512 +   0 .. 255] = Gr (row-major 16x16)
//                                [g*512 + 256 .. 511] = Gi
//                     [1536 + k] = zW[k], k in 0..4095
// ---------------------------------------------------------------------------
__global__ void qsim_setup(const float* __restrict__ params,
                           const float* __restrict__ W,
                           float* __restrict__ ws) {
  __shared__ float u_re[NW][2][2];
  __shared__ float u_im[NW][2][2];
  int t = threadIdx.x;
  if (t < NW) {
    float th = params[2 * t] * 0.5f;
    float ph = params[2 * t + 1] * 0.5f;
    float c = cosf(th), s = sinf(th);
    float cy = cosf(ph), sy = sinf(ph);
    // U = RY * RX, RX=[[c,-is],[-is,c]], RY=[[cy,-sy],[sy,cy]]
    u_re[t][0][0] = cy * c;   u_im[t][0][0] =  sy * s;
    u_re[t][0][1] = -sy * c;  u_im[t][0][1] = -cy * s;
    u_re[t][1][0] =  sy * c;  u_im[t][1][0] = -cy * s;
    u_re[t][1][1] =  cy * c;  u_im[t][1][1] = -sy * s;
  }
  __syncthreads();
  for (int e = t; e < 3 * 256; e += blockDim.x) {
    int g = e >> 8;
    int r = (e >> 4) & 15, c = e & 15;
    float ar = 1.f, ai = 0.f;
    for (int q = 0; q < 4; ++q) {
      int w = g * 4 + q;
      int rb = (r >> (3 - q)) & 1;
      int cb = (c >> (3 - q)) & 1;
      float br = u_re[w][rb][cb], bi = u_im[w][rb][cb];
      float nr = ar * br - ai * bi;
      float ni = ar * bi + ai * br;
      ar = nr; ai = ni;
    }
    ws[g * 512 + r * 16 + c]       = ar;
    ws[g * 512 + 256 + r * 16 + c] = ai;
  }
  // zW[k] = sum_j W_j * (1 - 2*parity(wire bits 0..j of k)); wire 0 = MSB.
  // (CNOT chain folded: post-CNOT bit j = prefix parity of pre-CNOT bits.)
  for (int k = t; k < QDIM; k += blockDim.x) {
    float acc = 0.f;
    int par = 0;
    for (int j = 0; j < NW; ++j) {
      par ^= (k >> (NW - 1 - j)) & 1;
      acc += W[j] * (par ? -1.f : 1.f);
    }
    ws[1536 + k] = acc;
  }
}

// ---------------------------------------------------------------------------
// Main: one state per block, 128 threads = 4 wave32.
// State tensor view: s[i][j][k], 16x16x16, flat idx = i*256 + j*16 + k.
// Group0 contracts i (input real), Group1 contracts j, Group2 contracts k.
// Each group = 16 disjoint 16x16x16 complex GEMM tiles on v_wmma_f32_16x16x4.
// Tile loops use a compile-time trip count of 4 (tile = 4*it + wave) so the
// compiler emits straight-line unrolled code with EXEC all-ones around WMMA.
// ---------------------------------------------------------------------------
__device__ __forceinline__ v8f wmma4(v2f a, v2f b, v8f c) {
  return __builtin_amdgcn_wmma_f32_16x16x4_f32(false, a, false, b,
                                               (short)0, c, false, false);
}

__global__ __launch_bounds__(128) void qsim_main(
    const float* __restrict__ sb, const float* __restrict__ ws,
    const float* __restrict__ bptr, float* __restrict__ out) {
  __shared__ float sr[QDIM];
  __shared__ float si[QDIM];
  __shared__ float red[4];

  const int tid  = threadIdx.x;
  const int lane = tid & 31;
  const int wave = tid >> 5;
  const int h    = lane >> 4;   // lane half
  const int l16  = lane & 15;
  const float* in = sb + (size_t)blockIdx.x * QDIM;

  for (int k = tid; k < QDIM; k += 128) sr[k] = in[k];
  __syncthreads();

  // ---- Group 0: out[i'][jk] = sum_i G0[i'][i] * s[i][jk]  (s real) ----
  {
    const float* Gr = ws;
    const float* Gi = ws + 256;
    v2f gr[4], gi[4];
#pragma unroll
    for (int c = 0; c < 4; ++c) {
      int col = 4 * c + 2 * h;
      gr[c] = v2f{Gr[l16 * 16 + col], Gr[l16 * 16 + col + 1]};
      gi[c] = v2f{Gi[l16 * 16 + col], Gi[l16 * 16 + col + 1]};
    }
#pragma unroll
    for (int it = 0; it < 4; ++it) {
      int tile = 4 * it + wave;
      v2f bs[4];
#pragma unroll
      for (int c = 0; c < 4; ++c) {
        int k0 = 4 * c + 2 * h;
        bs[c] = v2f{sr[(k0 + 0) * 256 + tile * 16 + l16],
                    sr[(k0 + 1) * 256 + tile * 16 + l16]};
      }
      v8f aRe = {}, aIm = {};
#pragma unroll
      for (int c = 0; c < 4; ++c) {
        aRe = wmma4(gr[c], bs[c], aRe);
        aIm = wmma4(gi[c], bs[c], aIm);
      }
#pragma unroll
      for (int r = 0; r < 8; ++r) {
        int M = r + 8 * h;
        sr[M * 256 + tile * 16 + l16] = aRe[r];
        si[M * 256 + tile * 16 + l16] = aIm[r];
      }
    }
  }
  __syncthreads();

  // ---- Group 1: per slab i: out_i = G1 x s_i (16x16 complex) ----
  {
    const float* Gr = ws + 512;
    const float* Gi = ws + 512 + 256;
    v2f gr[4], gi[4], gin[4];
#pragma unroll
    for (int c = 0; c < 4; ++c) {
      int col = 4 * c + 2 * h;
      gr[c]  = v2f{Gr[l16 * 16 + col], Gr[l16 * 16 + col + 1]};
      gi[c]  = v2f{Gi[l16 * 16 + col], Gi[l16 * 16 + col + 1]};
      gin[c] = v2f{-gi[c].x, -gi[c].y};   // f32 WMMA has no A/B neg -> prefold
    }
#pragma unroll
    for (int it = 0; it < 4; ++it) {
      int i = 4 * it + wave;
      v2f br[4], bi[4];
#pragma unroll
      for (int c = 0; c < 4; ++c) {
        int k0 = 4 * c + 2 * h;
        br[c] = v2f{sr[i * 256 + (k0 + 0) * 16 + l16],
                    sr[i * 256 + (k0 + 1) * 16 + l16]};
        bi[c] = v2f{si[i * 256 + (k0 + 0) * 16 + l16],
                    si[i * 256 + (k0 + 1) * 16 + l16]};
      }
      v8f aRe = {}, aIm = {};
#pragma unroll
      for (int c = 0; c < 4; ++c) {
        aRe = wmma4(gr[c],  br[c], aRe);
        aRe = wmma4(gin[c], bi[c], aRe);   // - Gi*Si
        aIm = wmma4(gi[c],  br[c], aIm);
        aIm = wmma4(gr[c],  bi[c], aIm);
      }
#pragma unroll
      for (int r = 0; r < 8; ++r) {
        int M = r + 8 * h;
        sr[i * 256 + M * 16 + l16] = aRe[r];
        si[i * 256 + M * 16 + l16] = aIm[r];
      }
    }
  }
  __syncthreads();

  // ---- Group 2: out[r][k'] = sum_k s[r][k]*G2[k'][k]  (A=state, B=G2^T) ----
  {
    const float* Gr = ws + 1024;
    const float* Gi = ws + 1024 + 256;
    v2f bgr[4], bgi[4], bgin[4];
#pragma unroll
    for (int c = 0; c < 4; ++c) {
      int col = 4 * c + 2 * h;                 // B[K][N] = G2[N][4c+K]
      bgr[c]  = v2f{Gr[l16 * 16 + col], Gr[l16 * 16 + col + 1]};
      bgi[c]  = v2f{Gi[l16 * 16 + col], Gi[l16 * 16 + col + 1]};
      bgin[c] = v2f{-bgi[c].x, -bgi[c].y};
    }
#pragma unroll
    for (int it = 0; it < 4; ++it) {
      int r0 = (4 * it + wave) * 16;
      v2f ar[4], ai[4];
#pragma unroll
      for (int c = 0; c < 4; ++c) {
        int col = 4 * c + 2 * h;               // A[M][K] = s[r0+M][4c+K]
        ar[c] = v2f{sr[(r0 + l16) * 16 + col], sr[(r0 + l16) * 16 + col + 1]};
        ai[c] = v2f{si[(r0 + l16) * 16 + col], si[(r0 + l16) * 16 + col + 1]};
      }
      v8f aRe = {}, aIm = {};
#pragma unroll
      for (int c = 0; c < 4; ++c) {
        aRe = wmma4(ar[c], bgr[c],  aRe);
        aRe = wmma4(ai[c], bgin[c], aRe);      // - Si*Gi
        aIm = wmma4(ar[c], bgi[c],  aIm);
        aIm = wmma4(ai[c], bgr[c],  aIm);
      }
#pragma unroll
      for (int r = 0; r < 8; ++r) {
        int M = r + 8 * h;
        sr[(r0 + M) * 16 + l16] = aRe[r];
        si[(r0 + M) * 16 + l16] = aIm[r];
      }
    }
  }
  __syncthreads();

  // ---- probs . zW reduction ----
  const float* zW = ws + 1536;
  float acc = 0.f;
  for (int k = tid; k < QDIM; k += 128) {
    float re = sr[k], im = si[k];
    acc += (re * re + im * im) * zW[k];
  }
#pragma unroll
  for (int off = 16; off; off >>= 1) acc += __shfl_down(acc, off, 32);
  if (lane == 0) red[wave] = acc;
  __syncthreads();
  if (tid == 0) out[blockIdx.x] = red[0] + red[1] + red[2] + red[3] + bptr[0];
}

extern "C" void kernel_launch(void* const* d_in, const int* in_sizes, int n_in,
                              void* d_out, int out_size, void* d_ws, size_t ws_size,
                              hipStream_t stream) {
  const float* state  = (const float*)d_in[0];  // [4096, 4096]
  const float* params = (const float*)d_in[1];  // [24]
  const float* W      = (const float*)d_in[2];  // [12]
  const float* b      = (const float*)d_in[3];  // [1]
  float* out = (float*)d_out;                   // [4096]
  float* ws  = (float*)d_ws;                    // needs 5632 floats

  qsim_setup<<<1, 256, 0, stream>>>(params, W, ws);
  qsim_main<<<4096, 128, 0, stream>>>(state, ws, b, out);
}